// SparseInterpolate2D_51015621542410
// MI455X (gfx1250) — compile-verified
//
#include <hip/hip_runtime.h>

// Problem constants (from reference)
constexpr int B_ = 4, H_ = 512, W_ = 512, C_ = 64;
constexpr float INV_VX = 5.0f;   // 1 / (102.4/512)
constexpr float INV_VY = 5.0f;   // 1 / (102.4/512)
constexpr float EPS_   = 1e-8f;

typedef unsigned int u32x4 __attribute__((ext_vector_type(4)));
typedef int          i32x4 __attribute__((ext_vector_type(4)));
typedef int          i32x8 __attribute__((ext_vector_type(8)));

// Tensor Data Mover: 1-D load of `td0` dwords (clamped for OOB-zero-fill) from
// `gaddr` into LDS offset `lds_off`; tile_dim0 = `tile0` dwords.
// Descriptor packing per CDNA5 ISA.
__device__ __forceinline__ void tdm_load_dwords_to_lds(unsigned lds_off,
                                                       unsigned long long gaddr,
                                                       unsigned td0, unsigned tile0) {
#if __has_builtin(__builtin_amdgcn_tensor_load_to_lds)
    u32x4 g0 = { 1u,                                   // count=1, user descriptor
                 lds_off,                              // lds_addr
                 (unsigned)gaddr,                      // global_addr[31:0]
                 ((unsigned)(gaddr >> 32) & 0x01FFFFFFu) | 0x80000000u }; // [56:32] | type=2
    i32x8 g1 = { (int)0x00020000u,                     // data_size=2 (4B), no flags
                 (int)((td0 & 0xFFFFu) << 16),         // tensor_dim0[15:0]
                 (int)((td0 >> 16) | (1u << 16)),      // tensor_dim0[31:16] | tensor_dim1[15:0]=1
                 (int)(tile0 << 16),                   // tensor_dim1[31:16]=0 | tile_dim0
                 0,                                    // tile_dim1=0, tile_dim2=0 (unused)
                 (int)td0,                             // tensor_dim0_stride[31:0]
                 0, 0 };                               // stride hi bits / dim1 stride
    i32x4 gz = { 0, 0, 0, 0 };                         // groups 2/3: dims >= 2 unused
#if defined(__clang_major__) && __clang_major__ >= 23
    i32x8 gz8 = { 0, 0, 0, 0, 0, 0, 0, 0 };
    __builtin_amdgcn_tensor_load_to_lds(g0, g1, gz, gz, gz8, 0);
#else
    __builtin_amdgcn_tensor_load_to_lds(g0, g1, gz, gz, 0);
#endif
    __builtin_amdgcn_s_wait_tensorcnt(0);
#else
    (void)lds_off; (void)gaddr; (void)td0; (void)tile0;
#endif
}

// ---------------------------------------------------------------------------
// Kernel 1: fill index grid with -1 (int4 vectorized, one int4 per thread)
// ---------------------------------------------------------------------------
__global__ __launch_bounds__(256) void sp2d_init_grid(int* __restrict__ grid, int ncells) {
    int n4 = ncells >> 2;
    int4* g4 = reinterpret_cast<int4*>(grid);
    for (int j = blockIdx.x * blockDim.x + threadIdx.x; j < n4;
         j += gridDim.x * blockDim.x) {
        g4[j] = make_int4(-1, -1, -1, -1);
    }
}

// ---------------------------------------------------------------------------
// Kernel 2: scatter point-id into its (b,y,x) cell. Duplicate cells: arbitrary
// winner (matches JAX .at[].set with duplicate indices). Occupancy == idx>=0.
// ---------------------------------------------------------------------------
__global__ __launch_bounds__(256) void sp2d_scatter_idx(const int* __restrict__ indices,
                                                        int* __restrict__ grid, int n) {
    int i = blockIdx.x * blockDim.x + threadIdx.x;
    if (i >= n) return;
    int b = indices[3 * i + 0];
    int y = indices[3 * i + 1];
    int x = indices[3 * i + 2];
    grid[((size_t)b * H_ + y) * W_ + x] = i;
}

// ---------------------------------------------------------------------------
// Kernel 3: bilinear gather + normalize.
// 16 points per block; 16 lanes cooperate on one point, lane owns channels
// [4*lane, 4*lane+3] so each neighbor-row read is a coalesced 256B burst.
// Block's 48 xyz dwords staged into LDS via the Tensor Data Mover.
// ---------------------------------------------------------------------------
__global__ __launch_bounds__(256) void sp2d_interp(const float* __restrict__ features,
                                                   const float* __restrict__ xyz,
                                                   const int*   __restrict__ cnt,
                                                   const int*   __restrict__ grid,
                                                   float*       __restrict__ out,
                                                   int m) {
    __shared__ float sxyz[48];          // 16 points x (x,y,z)

    const int p0 = blockIdx.x * 16;     // first point of this block

#if __has_builtin(__builtin_amdgcn_tensor_load_to_lds)
    if (threadIdx.x == 0) {
        int avail = (m - p0) * 3;                     // valid dwords remaining
        unsigned td0 = (unsigned)(avail < 48 ? avail : 48);
        tdm_load_dwords_to_lds((unsigned)(uintptr_t)&sxyz[0],
                               (unsigned long long)(uintptr_t)(xyz + (size_t)p0 * 3),
                               td0, 48u);
    }
#else
    if (threadIdx.x < 48) {
        int g = p0 * 3 + (int)threadIdx.x;
        sxyz[threadIdx.x] = (g < m * 3) ? xyz[g] : 0.0f;
    }
#endif
    __syncthreads();

    int t    = blockIdx.x * blockDim.x + threadIdx.x;
    int p    = t >> 4;        // query point id
    int lp   = p & 15;        // point index within block
    int lane = t & 15;        // channel-chunk id within the point
    if (p >= m) return;

    // batch id from xyz_batch_cnt prefix (B_ == 4, tiny scan)
    int b = 0, cum = cnt[0];
    while (p >= cum && b < B_ - 1) { ++b; cum += cnt[b]; }

    float gx = sxyz[3 * lp + 0] * INV_VX - 0.5f;   // LDS broadcast reads
    float gy = sxyz[3 * lp + 1] * INV_VY - 0.5f;
    float x0 = floorf(gx);
    float y0 = floorf(gy);

    const int* gbase = grid + (size_t)b * H_ * W_;

    // Pass 1: indices + weights for the 4 taps, prefetch feature rows early.
    float wv[4];
    int   fi[4];
#pragma unroll
    for (int dy = 0; dy < 2; ++dy) {
        float yn = y0 + (float)dy;
        float wy = fmaxf(1.0f - fabsf(gy - yn), 0.0f);
        bool  vy = (yn >= 0.0f) && (yn < (float)H_);
        int   yi = (int)yn; yi = yi < 0 ? 0 : (yi > H_ - 1 ? H_ - 1 : yi);
#pragma unroll
        for (int dx = 0; dx < 2; ++dx) {
            float xn = x0 + (float)dx;
            float wx = fmaxf(1.0f - fabsf(gx - xn), 0.0f);
            bool  vx = (xn >= 0.0f) && (xn < (float)W_);
            int   xi = (int)xn; xi = xi < 0 ? 0 : (xi > W_ - 1 ? W_ - 1 : xi);

            int   k    = dy * 2 + dx;
            float w    = wx * wy * ((vx && vy) ? 1.0f : 0.0f);
            int   fidx = gbase[yi * W_ + xi];
            wv[k] = w;
            fi[k] = fidx;
            if (fidx >= 0 && w > 0.0f) {
                // gfx1250 global_prefetch_b8: warm the 256B feature row
                __builtin_prefetch(features + (size_t)fidx * C_ + lane * 4, 0, 0);
            }
        }
    }

    // Pass 2: weighted accumulate (coalesced b128 loads, L2-resident).
    float4 acc = make_float4(0.0f, 0.0f, 0.0f, 0.0f);
    float  wsum = 0.0f;
#pragma unroll
    for (int k = 0; k < 4; ++k) {
        float w = wv[k];
        int   fidx = fi[k];
        if (fidx >= 0 && w > 0.0f) {
            const float4* row = reinterpret_cast<const float4*>(features + (size_t)fidx * C_);
            float4 v = row[lane];
            acc.x = fmaf(w, v.x, acc.x);
            acc.y = fmaf(w, v.y, acc.y);
            acc.z = fmaf(w, v.z, acc.z);
            acc.w = fmaf(w, v.w, acc.w);
            wsum += w;
        }
    }

    float inv = 1.0f / fmaxf(wsum, EPS_);
    float4 r  = make_float4(acc.x * inv, acc.y * inv, acc.z * inv, acc.w * inv);
    reinterpret_cast<float4*>(out)[(size_t)p * (C_ / 4) + lane] = r;
}

// ---------------------------------------------------------------------------
// Launch: inputs are [features (N*C f32), indices (N*3 i32), xyz (M*3 f32),
// xyz_batch_cnt (B i32)]. d_ws holds the 4 MiB index grid (rebuilt per call).
// ---------------------------------------------------------------------------
extern "C" void kernel_launch(void* const* d_in, const int* in_sizes, int n_in,
                              void* d_out, int out_size, void* d_ws, size_t ws_size,
                              hipStream_t stream) {
    const float* features = (const float*)d_in[0];
    const int*   indices  = (const int*)d_in[1];
    const float* xyz      = (const float*)d_in[2];
    const int*   cnt      = (const int*)d_in[3];
    float*       out      = (float*)d_out;
    int*         grid     = (int*)d_ws;   // B*H*W ints = 4 MiB

    const int n = in_sizes[1] / 3;        // N points with features
    const int m = in_sizes[2] / 3;        // M query points
    const int ncells = B_ * H_ * W_;      // 1,048,576

    // 1) index grid = -1  (exactly one int4 per thread: 1024*256*4 = ncells)
    sp2d_init_grid<<<1024, 256, 0, stream>>>(grid, ncells);
    // 2) scatter point ids
    sp2d_scatter_idx<<<(n + 255) / 256, 256, 0, stream>>>(indices, grid, n);
    // 3) gather + normalize (16 points per block, 16 lanes per point)
    const int blocks = (m + 15) / 16;
    sp2d_interp<<<blocks, 256, 0, stream>>>(features, xyz, cnt, grid, out, m);
}